// GaussRenderer_34342558499261
// MI455X (gfx1250) — compile-verified
//
#include <hip/hip_runtime.h>
#include <math.h>

// ---------------------------------------------------------------------------
// Problem constants (must match reference)
// ---------------------------------------------------------------------------
#define P_N    8192
#define W_IMG  192
#define H_IMG  192
#define TILE   16
#define NTX    (W_IMG / TILE)   // 12
#define NTY    (H_IMG / TILE)   // 12
#define CHUNK  256              // gaussians staged per LDS chunk
#define NGROUP (CHUNK / 4)      // 4-gaussian groups per chunk (K=4 per WMMA)

#define TANF   0.5773502691896257f          // tan(60deg/2)
#define FXY    (96.0f / 0.5773502691896257f) // fx = fy = cot(fov/2)*W/2

typedef float        v2f __attribute__((ext_vector_type(2)));
typedef float        v8f __attribute__((ext_vector_type(8)));
typedef unsigned int v4u __attribute__((ext_vector_type(4)));
typedef int          v8i __attribute__((ext_vector_type(8)));
typedef int          v4i __attribute__((ext_vector_type(4)));

#if __has_builtin(__builtin_amdgcn_tensor_load_to_lds) && \
    __has_builtin(__builtin_amdgcn_s_wait_tensorcnt)
#define USE_TDM 1
#endif

// ---------------------------------------------------------------------------
// Kernel 1: per-gaussian preprocessing
// Record layout (16 f32 per gaussian):
//  [0]mx [1]my [2]conic00 [3]conic01 [4]conic11 [5]opac(masked)
//  [6]rminx [7]rminy [8]rmaxx [9]rmaxy [10]depth [11]r [12]g [13]b [14]0 [15]0
// ---------------------------------------------------------------------------
__global__ __launch_bounds__(256) void gs_preprocess(
    const float* __restrict__ means3d, const float* __restrict__ scales,
    const float* __restrict__ rots,    const float* __restrict__ colors,
    const float* __restrict__ opacities, const float* __restrict__ vm,
    const float* __restrict__ pm, float* __restrict__ recU,
    float* __restrict__ radii_out)
{
  int i = blockIdx.x * blockDim.x + threadIdx.x;
  if (i >= P_N) return;

  float p0 = means3d[3*i+0], p1 = means3d[3*i+1], p2 = means3d[3*i+2];
  float sx = scales[3*i+0],  sy = scales[3*i+1],  sz = scales[3*i+2];
  float qw = rots[4*i+0], qx = rots[4*i+1], qy = rots[4*i+2], qz = rots[4*i+3];
  float qn = rsqrtf(qw*qw + qx*qx + qy*qy + qz*qz);
  qw *= qn; qx *= qn; qy *= qn; qz *= qn;

  float R00 = 1.f - 2.f*(qy*qy + qz*qz), R01 = 2.f*(qx*qy - qw*qz), R02 = 2.f*(qx*qz + qw*qy);
  float R10 = 2.f*(qx*qy + qw*qz), R11 = 1.f - 2.f*(qx*qx + qz*qz), R12 = 2.f*(qy*qz - qw*qx);
  float R20 = 2.f*(qx*qz - qw*qy), R21 = 2.f*(qy*qz + qw*qx), R22 = 1.f - 2.f*(qx*qx + qy*qy);
  float s0 = sx*sx, s1 = sy*sy, s2 = sz*sz;
  // Sigma = (R*diag(s)) (R*diag(s))^T
  float S00 = R00*R00*s0 + R01*R01*s1 + R02*R02*s2;
  float S01 = R00*R10*s0 + R01*R11*s1 + R02*R12*s2;
  float S02 = R00*R20*s0 + R01*R21*s1 + R02*R22*s2;
  float S11 = R10*R10*s0 + R11*R11*s1 + R12*R12*s2;
  float S12 = R10*R20*s0 + R11*R21*s1 + R12*R22*s2;
  float S22 = R20*R20*s0 + R21*R21*s1 + R22*R22*s2;

  // t = V [p;1]
  float t0 = vm[0]*p0 + vm[1]*p1 + vm[2]*p2 + vm[3];
  float t1 = vm[4]*p0 + vm[5]*p1 + vm[6]*p2 + vm[7];
  float t2 = vm[8]*p0 + vm[9]*p1 + vm[10]*p2 + vm[11];
  float invz = 1.f / t2;
  const float lim = 1.3f * TANF;
  float txc = fminf(fmaxf(t0*invz, -lim), lim) * t2;
  float tyc = fminf(fmaxf(t1*invz, -lim), lim) * t2;
  float j00 = FXY * invz, j02 = -txc * invz * invz * FXY;
  float j11 = FXY * invz, j12 = -tyc * invz * invz * FXY;

  // M = W Sigma W^T, W = view[:3,:3]
  float W00=vm[0],W01=vm[1],W02=vm[2];
  float W10=vm[4],W11=vm[5],W12=vm[6];
  float W20=vm[8],W21=vm[9],W22=vm[10];
  float U00 = W00*S00 + W01*S01 + W02*S02;
  float U01 = W00*S01 + W01*S11 + W02*S12;
  float U02 = W00*S02 + W01*S12 + W02*S22;
  float U10 = W10*S00 + W11*S01 + W12*S02;
  float U11 = W10*S01 + W11*S11 + W12*S12;
  float U12 = W10*S02 + W11*S12 + W12*S22;
  float U20 = W20*S00 + W21*S01 + W22*S02;
  float U21 = W20*S01 + W21*S11 + W22*S12;
  float U22 = W20*S02 + W21*S12 + W22*S22;
  float M00 = U00*W00 + U01*W01 + U02*W02;
  float M01 = U00*W10 + U01*W11 + U02*W12;
  float M11 = U10*W10 + U11*W11 + U12*W12;
  float M20 = U20*W00 + U21*W01 + U22*W02;
  float M21 = U20*W10 + U21*W11 + U22*W12;

  // cov2d = (J M J)[:2,:2]  (exactly as reference einsum: third factor is J)
  float cov00 = (j00*M00 + j02*M20) * j00;
  float cov01 = (j00*M01 + j02*M21) * j11;
  float cov10 = (j11*M01 + j12*M20) * j00;
  float cov11 = (j11*M11 + j12*M21) * j11;
  float det2 = cov00*cov11 - cov01*cov10;

  float mid = 0.5f * (cov00 + cov11);
  float sd  = sqrtf(fmaxf(mid*mid - det2, 1e-8f));
  float rad = ceilf(3.f * sqrtf(mid + sd));

  float det = fmaxf(det2, 0.001f);
  float idet = 1.f / det;
  float c00 = cov11 * idet, c01 = -cov01 * idet, c11 = cov00 * idet;

  // projection: rows 0,1,3 of (pm @ vm)
  float PV0[4], PV1[4], PV3[4];
  for (int j = 0; j < 4; ++j) {
    PV0[j] = pm[0]*vm[j]  + pm[1]*vm[4+j]  + pm[2]*vm[8+j]  + pm[3]*vm[12+j];
    PV1[j] = pm[4]*vm[j]  + pm[5]*vm[4+j]  + pm[6]*vm[8+j]  + pm[7]*vm[12+j];
    PV3[j] = pm[12]*vm[j] + pm[13]*vm[4+j] + pm[14]*vm[8+j] + pm[15]*vm[12+j];
  }
  float c0 = PV0[0]*p0 + PV0[1]*p1 + PV0[2]*p2 + PV0[3];
  float c1 = PV1[0]*p0 + PV1[1]*p1 + PV1[2]*p2 + PV1[3];
  float c3 = PV3[0]*p0 + PV3[1]*p1 + PV3[2]*p2 + PV3[3];
  float iw = 1.f / (c3 + 1e-6f);
  float mx = (c0*iw + 1.f) * 0.5f * (float)(W_IMG - 1);
  float my = (c1*iw + 1.f) * 0.5f * (float)(H_IMG - 1);
  float opv = opacities[i] * ((t2 >= 0.2f) ? 1.f : 0.f);

  float* r = recU + 16*i;
  r[0]=mx; r[1]=my; r[2]=c00; r[3]=c01; r[4]=c11; r[5]=opv;
  r[6]=fmaxf(mx - rad, 0.f);
  r[7]=fmaxf(my - rad, 0.f);
  r[8]=fminf(mx + rad, (float)(W_IMG - 1));
  r[9]=fminf(my + rad, (float)(H_IMG - 1));
  r[10]=t2;
  r[11]=colors[3*i+0]; r[12]=colors[3*i+1]; r[13]=colors[3*i+2];
  r[14]=0.f; r[15]=0.f;
  radii_out[i] = rad;
}

// ---------------------------------------------------------------------------
// Kernel 2: single-workgroup bitonic argsort of 8192 depths (keys in LDS)
// ---------------------------------------------------------------------------
__global__ __launch_bounds__(1024) void gs_sort(const float* __restrict__ recU,
                                                int* __restrict__ order)
{
  __shared__ float key[P_N];
  __shared__ int   idx[P_N];
  for (int i = threadIdx.x; i < P_N; i += 1024) { key[i] = recU[16*i + 10]; idx[i] = i; }
  __syncthreads();
  for (int k = 2; k <= P_N; k <<= 1) {
    for (int j = k >> 1; j > 0; j >>= 1) {
      for (int i = threadIdx.x; i < P_N; i += 1024) {
        int ixj = i ^ j;
        if (ixj > i) {
          bool up = ((i & k) == 0);
          float ki = key[i], kj = key[ixj];
          if (up ? (ki > kj) : (ki < kj)) {
            key[i] = kj; key[ixj] = ki;
            int t = idx[i]; idx[i] = idx[ixj]; idx[ixj] = t;
          }
        }
      }
      __syncthreads();
    }
  }
  for (int i = threadIdx.x; i < P_N; i += 1024) order[i] = idx[i];
}

// ---------------------------------------------------------------------------
// Kernel 3: gather sorted records + pre-swizzle features into the
// V_WMMA_F32_16X16X4_F32 B-matrix layout (4x16 tile per 4-gaussian group):
//   VGPR v, lane (n + 16*(r>=2)) holds feature n of group-gaussian r, v = r&1.
//   features: n=0..2 rgb, n=3 depth, n=4 one (alpha accumulation), else 0.
// ---------------------------------------------------------------------------
__global__ __launch_bounds__(256) void gs_gather(const float* __restrict__ recU,
    const int* __restrict__ order, float* __restrict__ geoS,
    float* __restrict__ BW)
{
  int i = blockIdx.x * blockDim.x + threadIdx.x;
  if (i >= P_N) return;
  int o = order[i];
  const float* r = recU + 16*o;
  float f[16];
  #pragma unroll
  for (int t = 0; t < 16; ++t) f[t] = r[t];
  float* g = geoS + 16*i;
  #pragma unroll
  for (int t = 0; t < 16; ++t) g[t] = f[t];

  int g4 = i >> 2, rr = i & 3;
  int laneBase = (rr >= 2) ? 16 : 0;
  int v = rr & 1;
  #pragma unroll
  for (int n = 0; n < 16; ++n) {
    float feat = (n == 0) ? f[11] : (n == 1) ? f[12] : (n == 2) ? f[13]
               : (n == 3) ? f[10] : (n == 4) ? 1.f : 0.f;
    BW[g4*64 + (laneBase + n)*2 + v] = feat;
  }
}

// ---------------------------------------------------------------------------
// Tensor Data Mover: 1D tile of CHUNK*16 dwords, global -> LDS.
// This toolchain exposes the 6-arg builtin:
//   (uint32x4 g0, int32x8 g1, int32x4 g2, int32x4 g3, int32x8 g4, i32 cpol)
// Groups 2/3 disabled (2D tensor path); trailing group zero-filled.
// ---------------------------------------------------------------------------
#ifdef USE_TDM
__device__ __forceinline__ void gs_tdm_issue(const float* gsrc, unsigned ldsAddr)
{
  unsigned long long ga = (unsigned long long)(const void*)gsrc;
  const int nE = CHUNK * 16;  // 4096 dwords = 16KB
  // D# group0: count=1 | lds_addr | global_addr[56:0] | type=2
  v4u g0 = { 1u, ldsAddr, (unsigned)ga,
             (unsigned)((ga >> 32) & 0x01FFFFFFull) | 0x80000000u };
  // D# group1: data_size=4B; tensor_dim0=nE; tensor_dim1=1; tile_dim0=nE;
  //            tile_dim1/2=0 (1D); tensor_dim0_stride=nE
  v8i g1 = { (int)(2u << 16),
             (int)((unsigned)(nE & 0xFFFF) << 16),
             (int)(((unsigned)nE >> 16) | (1u << 16)),
             (int)((unsigned)(nE & 0xFFFF) << 16),
             0,
             nE,
             0, 0 };
  v4i gz4 = { 0, 0, 0, 0 };
  v8i gz8 = { 0, 0, 0, 0, 0, 0, 0, 0 };
  __builtin_amdgcn_tensor_load_to_lds(g0, g1, gz4, gz4, gz8, 0);
}
#endif

// ---------------------------------------------------------------------------
// per-lane alpha evaluation (one gaussian, one pixel)
// ---------------------------------------------------------------------------
__device__ __forceinline__ float gs_alpha(const float* __restrict__ rec,
                                          float fpx, float fpy,
                                          float tx0, float ty0)
{
  float mx = rec[0], my = rec[1];
  float c00 = rec[2], c01 = rec[3], c11 = rec[4], op = rec[5];
  float tlx = fmaxf(rec[6], tx0), tly = fmaxf(rec[7], ty0);
  float brx = fminf(rec[8], tx0 + (float)(TILE - 1));
  float bry = fminf(rec[9], ty0 + (float)(TILE - 1));
  float msk = (brx > tlx && bry > tly) ? 1.f : 0.f;
  float dx = fpx - mx, dy = fpy - my;
  float e = -0.5f * (dx*dx*c00 + dy*dy*c11) - dx*dy*c01;
  float gw = __expf(e);
  return fminf(gw * op, 0.99f) * msk;
}

// ---------------------------------------------------------------------------
// Kernel 4: tiled alpha-blend render.
// Block = 1 tile, 512 threads = 16 waves. Wave w owns tile row w (16 pixels =
// the 16 M-rows of a 16x16 WMMA tile, N-columns = 5 features).
// A-matrix (16x4 f32): lanes 0-15 K={0,1}, lanes 16-31 K={2,3} per ISA layout.
// Transmittance kept per-pixel, duplicated in lanes m and m+16, exchanged
// between halves with shfl_xor(16).
// ---------------------------------------------------------------------------
__global__ __launch_bounds__(512) void gs_render(
    const float* __restrict__ geoS, const float* __restrict__ BW,
    float* __restrict__ out)
{
  __shared__ float sh_geo[2][CHUNK * 16];  // 2 x 16KB
  __shared__ float sh_B[2][CHUNK * 16];    // 2 x 16KB

  int tx = blockIdx.x % NTX, ty = blockIdx.x / NTX;
  int lane = threadIdx.x & 31;
  int wv = threadIdx.x >> 5;
  int hi = lane >> 4;

  float fpx = (float)(tx*TILE + (lane & 15));
  float fpy = (float)(ty*TILE + wv);
  float tx0 = (float)(tx*TILE), ty0 = (float)(ty*TILE);

  float T = 1.f;
  v8f acc = {0.f,0.f,0.f,0.f,0.f,0.f,0.f,0.f};
  const int nChunks = P_N / CHUNK;

#ifdef USE_TDM
  if (wv == 0) {
    gs_tdm_issue(geoS, (unsigned)(unsigned long long)&sh_geo[0][0]);
    gs_tdm_issue(BW,   (unsigned)(unsigned long long)&sh_B[0][0]);
  }
#endif

  for (int c = 0; c < nChunks; ++c) {
    int buf = c & 1;
#ifdef USE_TDM
    if (wv == 0) {
      if (c + 1 < nChunks) {
        gs_tdm_issue(geoS + (size_t)(c+1)*CHUNK*16,
                     (unsigned)(unsigned long long)&sh_geo[buf ^ 1][0]);
        gs_tdm_issue(BW + (size_t)(c+1)*CHUNK*16,
                     (unsigned)(unsigned long long)&sh_B[buf ^ 1][0]);
        __builtin_amdgcn_s_wait_tensorcnt(2);  // chunk c fully landed
      } else {
        __builtin_amdgcn_s_wait_tensorcnt(0);
      }
    }
    __syncthreads();
#else
    for (int t = threadIdx.x; t < CHUNK*16; t += blockDim.x) {
      sh_geo[buf][t] = geoS[(size_t)c*CHUNK*16 + t];
      sh_B[buf][t]   = BW[(size_t)c*CHUNK*16 + t];
    }
    __syncthreads();
#endif

    for (int g = 0; g < NGROUP; ++g) {
      const float* recA = &sh_geo[buf][(g*4 + hi*2) * 16];
      float aA = gs_alpha(recA,      fpx, fpy, tx0, ty0);
      float aB = gs_alpha(recA + 16, fpx, fpy, tx0, ty0);
      float pSelf  = (1.f - aA) * (1.f - aB);
      float pOther = __shfl_xor(pSelf, 16, 32);
      float pre = hi ? pOther : 1.f;        // hi half needs (1-a0)(1-a1) prefix
      float wA = aA * T * pre;
      float wB = aB * T * pre * (1.f - aA);
      T *= pSelf * pOther;

      v2f av; av.x = wA; av.y = wB;
      const float* bp = &sh_B[buf][g*64 + lane*2];
      v2f bv; bv.x = bp[0]; bv.y = bp[1];

#if __has_builtin(__builtin_amdgcn_wmma_f32_16x16x4_f32)
      acc = __builtin_amdgcn_wmma_f32_16x16x4_f32(
          false, av, false, bv, (short)0, acc, false, false);
#else
      // fallback (keeps compilation alive; layout-approximate)
      #pragma unroll
      for (int q = 0; q < 8; ++q) acc[q] += av.x * bv.x + av.y * bv.y;
#endif
    }
    __syncthreads();
  }

  // D layout: VGPR v, lanes 0-15 -> M=v, lanes 16-31 -> M=v+8; N = lane&15
  int n = lane & 15;
  float* out_col = out;
  float* out_dep = out + (size_t)W_IMG * H_IMG * 3;
  float* out_acc = out + (size_t)W_IMG * H_IMG * 4;
  #pragma unroll
  for (int v = 0; v < 8; ++v) {
    int m = v + hi * 8;
    int px = tx*TILE + m;
    int py = ty*TILE + wv;
    int pix = py * W_IMG + px;
    float val = acc[v];
    if (n < 3)        out_col[pix*3 + n] = val;
    else if (n == 3)  out_dep[pix] = val;
    else if (n == 4)  out_acc[pix] = val;
  }
}

// ---------------------------------------------------------------------------
// Launcher
// ---------------------------------------------------------------------------
extern "C" void kernel_launch(void* const* d_in, const int* in_sizes, int n_in,
                              void* d_out, int out_size, void* d_ws, size_t ws_size,
                              hipStream_t stream)
{
  (void)in_sizes; (void)n_in; (void)out_size; (void)ws_size;
  const float* means3d   = (const float*)d_in[0];
  const float* scales    = (const float*)d_in[1];
  const float* rots      = (const float*)d_in[2];
  const float* colors    = (const float*)d_in[3];
  const float* opacities = (const float*)d_in[4];
  const float* vm        = (const float*)d_in[5];
  const float* pm        = (const float*)d_in[6];
  float* out = (float*)d_out;

  char* ws = (char*)d_ws;
  float* recU  = (float*)(ws + 0);                 // P*16 f32 = 512KB
  float* geoS  = (float*)(ws + 512*1024);          // P*16 f32 = 512KB (sorted)
  float* BW    = (float*)(ws + 1024*1024);         // P*16 f32 = 512KB (B layout)
  int*   order = (int*)  (ws + 1536*1024);         // P i32    = 32KB
  float* radii_out = out + (size_t)W_IMG * H_IMG * 5;

  gs_preprocess<<<P_N/256, 256, 0, stream>>>(means3d, scales, rots, colors,
                                             opacities, vm, pm, recU, radii_out);
  gs_sort<<<1, 1024, 0, stream>>>(recU, order);
  gs_gather<<<P_N/256, 256, 0, stream>>>(recU, order, geoS, BW);
  gs_render<<<NTX*NTY, 512, 0, stream>>>(geoS, BW, out);
}